// Correlation1d_84086869721444
// MI455X (gfx1250) — compile-verified
//
#include <hip/hip_runtime.h>

// CDNA5 / gfx1250 wave32 WMMA kernel for FlowNetC Correlation1d.
// out[b,i,h,w] = (1/C) * sum_c in1[b,c,h,w] * in2[b,c,h,w + 2i - 20]
//
// Stride-2 displacements decouple by parity of w: with w = 2u+r,
//   out_r[i,u] = sum_c A_r[c,u] * B_r[c, u+i-10],
//   A_r[c,u] = in1[..,2u+r], B_r[c,n] = in2[..,2n+r] (zero for n outside [0,64)).
// This is a banded Gram matrix G[m,n] (band n-m in [-10,10]) computed with
// V_WMMA_F32_16X16X4_F32 (full fp32). Per 16-row M-tile, three N-tiles at
// offsets {-16, 0, +16} cover the band. Out-of-range B columns are loaded as
// harmless garbage (clamped address) and replaced by literal 0 at store time,
// keeping the inner loop free of per-iteration cndmasks.

typedef float v2f __attribute__((ext_vector_type(2)));
typedef float v8f __attribute__((ext_vector_type(8)));

constexpr int kC  = 256;
constexpr int kH  = 64;
constexpr int kW  = 128;
constexpr int kND = 21;          // 2*(20/2)+1 displacement channels
constexpr int kHW = kH * kW;

__global__ __launch_bounds__(256) void corr1d_wmma_kernel(
    const float* __restrict__ in1,
    const float* __restrict__ in2,
    float* __restrict__ out)
{
    const int bh   = blockIdx.x;          // one block per (b, h)
    const int b    = bh / kH;
    const int h    = bh % kH;
    const int wv   = threadIdx.x >> 5;    // wave id 0..7
    const int lane = threadIdx.x & 31;
    const int half = lane >> 4;           // 0 or 1 (lane half, selects K pair)
    const int l16  = lane & 15;           // matrix row/col within tile
    const int m0   = (wv >> 1) << 4;      // M-tile base in parity space: 0,16,32,48
    const int r    = wv & 1;              // parity of w

    const size_t rowbase = ((size_t)b * kC * kH + (size_t)h) * kW; // (b, c=0, h, w=0)
    const float* p1 = in1 + rowbase;
    const float* p2 = in2 + rowbase;

    // A operand: A[m,k] = in1[b, c=k, h, 2*(m0+m)+r]; lane holds m=l16,
    // k = 2*half + {0,1} in (v0, v1) per the 32-bit 16x4 A layout.
    const int w1 = 2 * (m0 + l16) + r;

    // B operands for the 3 N-tiles at n0 = m0-16, m0, m0+16.
    // B[k,n] = in2[b, c=k, h, 2*n+r] for n in [0,64); OOB lanes clamp the
    // address (finite garbage, discarded at store).
    int  w2[3];
    bool vld[3];
    #pragma unroll
    for (int j = 0; j < 3; ++j) {
        const int n = m0 + (j - 1) * 16 + l16;
        vld[j] = (n >= 0) && (n < kW / 2);
        w2[j]  = vld[j] ? (2 * n + r) : 0;
    }

    const size_t cOff = (size_t)(2 * half) * kHW;
    const float* aP  = p1 + cOff + w1;
    const float* bP0 = p2 + cOff + w2[0];
    const float* bP1 = p2 + cOff + w2[1];
    const float* bP2 = p2 + cOff + w2[2];

    v8f acc0 = {0.f, 0.f, 0.f, 0.f, 0.f, 0.f, 0.f, 0.f};
    v8f acc1 = {0.f, 0.f, 0.f, 0.f, 0.f, 0.f, 0.f, 0.f};
    v8f acc2 = {0.f, 0.f, 0.f, 0.f, 0.f, 0.f, 0.f, 0.f};

    #pragma unroll 8
    for (int t = 0; t < kC / 4; ++t) {
        v2f a, b0, b1, b2;
        a.x  = aP[0];    a.y  = aP[kHW];
        b0.x = bP0[0];   b0.y = bP0[kHW];
        b1.x = bP1[0];   b1.y = bP1[kHW];
        b2.x = bP2[0];   b2.y = bP2[kHW];

        acc0 = __builtin_amdgcn_wmma_f32_16x16x4_f32(false, a, false, b0,
                                                     (short)0, acc0, false, false);
        acc1 = __builtin_amdgcn_wmma_f32_16x16x4_f32(false, a, false, b1,
                                                     (short)0, acc1, false, false);
        acc2 = __builtin_amdgcn_wmma_f32_16x16x4_f32(false, a, false, b2,
                                                     (short)0, acc2, false, false);

        aP  += (size_t)4 * kHW;
        bP0 += (size_t)4 * kHW;
        bP1 += (size_t)4 * kHW;
        bP2 += (size_t)4 * kHW;
    }

    // Store the band: D[m,n] -> out[b, i = n-m+10, h, w = 2m+r].
    // C/D layout: reg k holds row m0 + k + 8*half, lane l16 holds col n0 + l16.
    // Invalid columns (clamped loads) store literal zero, matching the
    // reference's zero padding.
    const float scale = 1.0f / (float)kC;
    float* po = out + (size_t)b * kND * kHW + (size_t)h * kW;

    #pragma unroll
    for (int j = 0; j < 3; ++j) {
        const v8f acc = (j == 0) ? acc0 : ((j == 1) ? acc1 : acc2);
        const int ncol = m0 + (j - 1) * 16 + l16;
        #pragma unroll
        for (int k = 0; k < 8; ++k) {
            const int mrow = m0 + k + 8 * half;
            const int i    = ncol - mrow + 10;
            if (i >= 0 && i < kND) {
                const float val = vld[j] ? acc[k] * scale : 0.0f;
                po[(size_t)i * kHW + 2 * mrow + r] = val;
            }
        }
    }
}

extern "C" void kernel_launch(void* const* d_in, const int* in_sizes, int n_in,
                              void* d_out, int out_size, void* d_ws, size_t ws_size,
                              hipStream_t stream) {
    const float* in1 = (const float*)d_in[0];
    const float* in2 = (const float*)d_in[1];
    float* out = (float*)d_out;

    const int B = in_sizes[0] / (kC * kH * kW);   // 8 for the reference shapes
    dim3 grid(B * kH);                            // one block per (b, h)
    dim3 block(256);                              // 8 wave32 waves
    corr1d_wmma_kernel<<<grid, block, 0, stream>>>(in1, in2, out);
}